// APPNPLayer_20667382628948
// MI455X (gfx1250) — compile-verified
//
#include <hip/hip_runtime.h>
#include <stdint.h>
#include <stddef.h>

// ---------------------------------------------------------------------------
// APPNP-style propagation on MI455X (gfx1250, wave32).
//   e = d[dst]*d[src];  10x: feat = r*((1-a)*SpMM(e,feat) + a*h) + (1-r)*feat
// L2-resident workload (features 25.6MB + edges ~7MB << 192MB L2).
// Strategy: build CSR-by-dst once (no fp32 atomics in the hot loop), then one
// wave per destination node, float4 per lane (128 dims / 32 lanes), 4x
// unrolled gather for >=4 loads in flight per wave.
// CDNA5 paths: TDM tensor_load_to_lds prefetch of h/feat rows into LDS
// (overlapped with the gather loop, s_wait_tensorcnt before use) and
// global_prefetch_b8 on upcoming gather rows.
// ---------------------------------------------------------------------------

#define N_NODES   50000
#define N_EDGES   600000
#define DIM       128
#define HOPS      10
#define ALPHA     0.1f

#define SCAN_TILE 512
#define WPB       8            // waves per block
#define TPB       (WPB * 32)

#if defined(__gfx1250__) && __has_builtin(__builtin_amdgcn_tensor_load_to_lds) && __has_builtin(__builtin_amdgcn_s_wait_tensorcnt)
#define USE_TDM 1
#else
#define USE_TDM 0
#endif

#if USE_TDM
typedef unsigned int v4u __attribute__((ext_vector_type(4)));
typedef int          v8i __attribute__((ext_vector_type(8)));
typedef int          v4i __attribute__((ext_vector_type(4)));

// Issue one TDM 1-D tile load: DIM fp32 from gsrc -> LDS at lds_dst.
// D# group0: count=1 | lds_addr | global_addr[56:0] | type=2 ("image").
// D# group1: data_size=4B, tensor_dim0=DIM, tensor_dim1=1, tile_dim0=DIM,
//            tile_dim1=0 (1-D), tensor_dim0_stride=DIM.
__device__ __forceinline__ void tdm_load_row(const float* gsrc, void* lds_dst) {
  unsigned lds = (unsigned)(uintptr_t)lds_dst;          // flat shared addr low 32b == LDS offset
  unsigned long long ga = (unsigned long long)(uintptr_t)gsrc;
  v4u g0;
  g0[0] = 1u;                                           // count=1, is_restore=0, gather off
  g0[1] = lds;                                          // lds_addr
  g0[2] = (unsigned)(ga & 0xFFFFFFFFu);                 // global_addr[31:0]
  g0[3] = (unsigned)((ga >> 32) & 0x01FFFFFFu) | (2u << 30); // addr[56:32] | type=2
  v8i g1;
  g1[0] = (2 << 16);                                    // workgroup_mask=0, data_size=2 (4B)
  g1[1] = (DIM & 0xFFFF) << 16;                         // tensor_dim0 low16 (bits 63:48)
  g1[2] = (1 << 16);                                    // tensor_dim0 hi=0, tensor_dim1=1
  g1[3] = (DIM & 0xFFFF) << 16;                         // tile_dim0 = DIM (bits 127:112)
  g1[4] = 0;                                            // tile_dim1=0 (1-D), tile_dim2=0
  g1[5] = DIM;                                          // tensor_dim0_stride low32
  g1[6] = 0;
  g1[7] = 0;
  v4i z4 = (v4i)(0);
  v8i z8 = (v8i)(0);
  // amdgpu-toolchain (clang-23 / therock-10.0) 6-arg form:
  //   (uint32x4 g0, int32x8 g1, int32x4 g2, int32x4 g3, int32x8 extra, i32 cpol)
  __builtin_amdgcn_tensor_load_to_lds(g0, g1, z4, z4, z8, 0);
}
#endif

// ----------------------------- CSR build -----------------------------------

__global__ void k_zero_i32(int* p, int n) {
  int i = blockIdx.x * blockDim.x + threadIdx.x;
  if (i < n) p[i] = 0;
}

__global__ void k_edge_w(const float* __restrict__ d, const int* __restrict__ src,
                         const int* __restrict__ dst, float* __restrict__ ew0,
                         int* __restrict__ counts) {
  int j = blockIdx.x * blockDim.x + threadIdx.x;
  if (j < N_EDGES) {
    int s = src[j], t = dst[j];
    ew0[j] = d[s] * d[t];
    atomicAdd(&counts[t], 1);
  }
}

__global__ void k_scan_tile(const int* __restrict__ counts, int* __restrict__ rowp,
                            int* __restrict__ tileSums, int n) {
  __shared__ int sh[SCAN_TILE];
  int t = threadIdx.x;
  int g = blockIdx.x * SCAN_TILE + t;
  int v = (g < n) ? counts[g] : 0;
  sh[t] = v;
  __syncthreads();
  for (int off = 1; off < SCAN_TILE; off <<= 1) {   // Hillis-Steele inclusive
    int x = (t >= off) ? sh[t - off] : 0;
    __syncthreads();
    sh[t] += x;
    __syncthreads();
  }
  int incl = sh[t];
  if (g < n) rowp[g] = incl - v;                    // exclusive prefix within tile
  if (t == SCAN_TILE - 1) tileSums[blockIdx.x] = incl;
}

__global__ void k_scan_sums(int* tileSums, int numTiles) {
  if (blockIdx.x == 0 && threadIdx.x == 0) {
    int acc = 0;
    for (int i = 0; i < numTiles; ++i) { int v = tileSums[i]; tileSums[i] = acc; acc += v; }
  }
}

__global__ void k_add_off(int* __restrict__ rowp, const int* __restrict__ tileSums,
                          int* __restrict__ fill, int n) {
  int i = blockIdx.x * blockDim.x + threadIdx.x;
  if (i < n) {
    int v = rowp[i] + tileSums[i / SCAN_TILE];
    rowp[i] = v;
    fill[i] = v;
  }
  if (i == n) rowp[n] = N_EDGES;
}

__global__ void k_scatter(const int* __restrict__ src, const int* __restrict__ dst,
                          const float* __restrict__ ew0, int* __restrict__ fill,
                          int* __restrict__ col, float* __restrict__ ew) {
  int j = blockIdx.x * blockDim.x + threadIdx.x;
  if (j < N_EDGES) {
    int t = dst[j];
    int pos = atomicAdd(&fill[t], 1);
    col[pos] = src[j];
    ew[pos]  = ew0[j];
  }
}

// ------------------------- fused SpMM + blend ------------------------------
// One wave per destination node. Lane l owns dims [4l, 4l+4): a gathered row
// is one coalesced 512B transaction. TDM prefetches h[v], feat_in[v] into LDS
// while the gather loop runs; s_wait_tensorcnt(0) before the blend. Inner
// broadcast loop unrolled 4x so >=4 independent gather loads are in flight.

__global__ __launch_bounds__(TPB) void
k_spmm(const float* __restrict__ feat_in, float* __restrict__ feat_out,
       const float* __restrict__ h, const int* __restrict__ col,
       const float* __restrict__ ew, const int* __restrict__ rowp,
       const float* __restrict__ layer_reg, int hop) {
#if USE_TDM
  __shared__ float sh_h[WPB][DIM];
  __shared__ float sh_f[WPB][DIM];
#endif
  const int lane = threadIdx.x & 31;
  const int wid  = threadIdx.x >> 5;
  const int v    = blockIdx.x * WPB + wid;
  if (v >= N_NODES) return;

#if USE_TDM
  // Kick off async row prefetch (TENSORcnt += 2), overlapped with gather loop.
  tdm_load_row(h       + (size_t)v * DIM, &sh_h[wid][0]);
  tdm_load_row(feat_in + (size_t)v * DIM, &sh_f[wid][0]);
#endif

  // Hoist scalar loads so their latency hides under the gather loop.
  const float r  = layer_reg[hop];
  const int rs = rowp[v];
  const int re = rowp[v + 1];

  float4 acc = make_float4(0.f, 0.f, 0.f, 0.f);
  for (int base = rs; base < re; base += 32) {
    int j = base + lane;
    int   c = 0;
    float w = 0.f;
    if (j < re) {
      c = col[j];
      w = ew[j];
      __builtin_prefetch(feat_in + (size_t)c * DIM, 0, 0);  // global_prefetch_b8
    }
    int cnt = re - base; if (cnt > 32) cnt = 32;

    int k = 0;
    for (; k + 4 <= cnt; k += 4) {
      // 4 independent broadcasts + 4 independent row loads in flight
      int   c0 = __shfl(c, k + 0, 32), c1 = __shfl(c, k + 1, 32);
      int   c2 = __shfl(c, k + 2, 32), c3 = __shfl(c, k + 3, 32);
      float w0 = __shfl(w, k + 0, 32), w1 = __shfl(w, k + 1, 32);
      float w2 = __shfl(w, k + 2, 32), w3 = __shfl(w, k + 3, 32);
      const float4 f0 = *((const float4*)(feat_in + (size_t)c0 * DIM) + lane);
      const float4 f1 = *((const float4*)(feat_in + (size_t)c1 * DIM) + lane);
      const float4 f2 = *((const float4*)(feat_in + (size_t)c2 * DIM) + lane);
      const float4 f3 = *((const float4*)(feat_in + (size_t)c3 * DIM) + lane);
      acc.x += w0 * f0.x; acc.y += w0 * f0.y; acc.z += w0 * f0.z; acc.w += w0 * f0.w;
      acc.x += w1 * f1.x; acc.y += w1 * f1.y; acc.z += w1 * f1.z; acc.w += w1 * f1.w;
      acc.x += w2 * f2.x; acc.y += w2 * f2.y; acc.z += w2 * f2.z; acc.w += w2 * f2.w;
      acc.x += w3 * f3.x; acc.y += w3 * f3.y; acc.z += w3 * f3.z; acc.w += w3 * f3.w;
    }
    for (; k < cnt; ++k) {
      int   ck = __shfl(c, k, 32);
      float wk = __shfl(w, k, 32);
      const float4 f = *((const float4*)(feat_in + (size_t)ck * DIM) + lane);
      acc.x += wk * f.x; acc.y += wk * f.y;
      acc.z += wk * f.z; acc.w += wk * f.w;
    }
  }

#if USE_TDM
  __builtin_amdgcn_s_wait_tensorcnt(0);
  const float4 hv = *((const float4*)&sh_h[wid][0] + lane);
  const float4 fv = *((const float4*)&sh_f[wid][0] + lane);
#else
  const float4 hv = *((const float4*)(h       + (size_t)v * DIM) + lane);
  const float4 fv = *((const float4*)(feat_in + (size_t)v * DIM) + lane);
#endif

  const float ca = (1.0f - ALPHA) * r;   // on agg
  const float ch = ALPHA * r;            // on h
  const float cf = 1.0f - r;             // on previous feat
  float4 o;
  o.x = ca * acc.x + ch * hv.x + cf * fv.x;
  o.y = ca * acc.y + ch * hv.y + cf * fv.y;
  o.z = ca * acc.z + ch * hv.z + cf * fv.z;
  o.w = ca * acc.w + ch * hv.w + cf * fv.w;
  *((float4*)(feat_out + (size_t)v * DIM) + lane) = o;
}

// ------------------------------ launcher -----------------------------------

extern "C" void kernel_launch(void* const* d_in, const int* in_sizes, int n_in,
                              void* d_out, int out_size, void* d_ws, size_t ws_size,
                              hipStream_t stream) {
  const float* h    = (const float*)d_in[0];
  const float* dd   = (const float*)d_in[1];
  const float* lreg = (const float*)d_in[2];
  const int*   src  = (const int*)d_in[3];
  const int*   dst  = (const int*)d_in[4];
  float* out = (float*)d_out;

  // carve workspace (512B aligned slices)
  char* w = (char*)d_ws;
  auto carve = [&](size_t bytes) -> char* {
    char* p = w;
    w += (bytes + 511) & ~(size_t)511;
    return p;
  };
  int*   counts   = (int*)  carve(sizeof(int)   * N_NODES);
  int*   rowp     = (int*)  carve(sizeof(int)   * (N_NODES + 1));
  int*   fill     = (int*)  carve(sizeof(int)   * N_NODES);
  int*   tileSums = (int*)  carve(sizeof(int)   * 128);
  float* ew0      = (float*)carve(sizeof(float) * N_EDGES);
  int*   colx     = (int*)  carve(sizeof(int)   * N_EDGES);
  float* eww      = (float*)carve(sizeof(float) * N_EDGES);
  float* feat_ws  = (float*)carve(sizeof(float) * (size_t)N_NODES * DIM);

  const int numTiles = (N_NODES + SCAN_TILE - 1) / SCAN_TILE;   // 98

  // 1) CSR build (re-done every call: deterministic, graph-capture safe)
  k_zero_i32<<<(N_NODES + 255) / 256, 256, 0, stream>>>(counts, N_NODES);
  k_edge_w  <<<(N_EDGES + 255) / 256, 256, 0, stream>>>(dd, src, dst, ew0, counts);
  k_scan_tile<<<numTiles, SCAN_TILE, 0, stream>>>(counts, rowp, tileSums, N_NODES);
  k_scan_sums<<<1, 64, 0, stream>>>(tileSums, numTiles);
  k_add_off <<<(N_NODES + 1 + 255) / 256, 256, 0, stream>>>(rowp, tileSums, fill, N_NODES);
  k_scatter <<<(N_EDGES + 255) / 256, 256, 0, stream>>>(src, dst, ew0, fill, colx, eww);

  // 2) 10 propagation hops; ping-pong so the last hop writes d_out
  const int blocks = (N_NODES + WPB - 1) / WPB;   // 6250
  const float* fin = h;
  for (int i = 0; i < HOPS; ++i) {
    float* fout = (i % 2 == 0) ? feat_ws : out;   // hop 9 (odd) -> out
    k_spmm<<<blocks, TPB, 0, stream>>>(fin, fout, h, colx, eww, rowp, lreg, i);
    fin = fout;
  }
}